// targeted_weight_dropout_12017318494849
// MI455X (gfx1250) — compile-verified
//
#include <hip/hip_runtime.h>

// Problem constants from the reference (w: (2048,1024,3,3) fp32; rand_u: (K,O)).
#define O_DIM   2048
#define K_DIM   9216            // 1024*3*3
#define THREADS 256
#define PER_THR (K_DIM / THREADS)   // 36 keys per thread
#define VEC4    (K_DIM / 4 / THREADS) // 9 float4 loads per thread
#define RANK    4608            // int(0.5 * 9216), 0-indexed order statistic

// ---- CDNA5 async global->LDS path (ASYNCcnt), guarded so we compile either way.
#if defined(__has_builtin)
#if __has_builtin(__builtin_amdgcn_global_load_async_to_lds_b32) && \
    __has_builtin(__builtin_amdgcn_s_wait_asynccnt)
#define USE_ASYNC_LDS 1
#endif
#endif
#ifndef USE_ASYNC_LDS
#define USE_ASYNC_LDS 0
#endif

typedef __attribute__((address_space(1))) int* gas1_i32;
typedef __attribute__((address_space(3))) int* gas3_i32;

// ---------------------------------------------------------------------------
// Kernel 1: per-row radix select of the RANK-th smallest |w[o,:]|.
// One block per row o; 36 keys/thread held in VGPRs; 4 MSB-first passes over
// a 256-bin LDS histogram (ds_add_u32) + Hillis-Steele scan.
// ---------------------------------------------------------------------------
__global__ __launch_bounds__(THREADS) void tw_threshold_kernel(
    const float* __restrict__ w, unsigned* __restrict__ thr_bits) {
  const int o = blockIdx.x;
  const int t = threadIdx.x;
  const float4* __restrict__ row4 =
      (const float4*)(w + (size_t)o * K_DIM);  // row base is 16B aligned

  unsigned keys[PER_THR];
#pragma unroll
  for (int j = 0; j < VEC4; ++j) {
    float4 v = row4[t + j * THREADS];          // global_load_b128, coalesced
    keys[4 * j + 0] = __float_as_uint(v.x) & 0x7FFFFFFFu;
    keys[4 * j + 1] = __float_as_uint(v.y) & 0x7FFFFFFFu;
    keys[4 * j + 2] = __float_as_uint(v.z) & 0x7FFFFFFFu;
    keys[4 * j + 3] = __float_as_uint(v.w) & 0x7FFFFFFFu;
  }

  __shared__ unsigned hist[THREADS];
  __shared__ unsigned scan[THREADS];
  __shared__ unsigned sel_bin;
  __shared__ unsigned sel_excl;

  unsigned prefix = 0u;   // high bytes of the target value found so far
  unsigned rank = RANK;   // rank within the surviving bucket

  for (int p = 3; p >= 0; --p) {
    hist[t] = 0u;
    __syncthreads();
    const int shift = p * 8;
#pragma unroll
    for (int i = 0; i < PER_THR; ++i) {
      const unsigned key = keys[i];
      // 64-bit shift: at p==3 this is key>>32 == 0 == prefix -> always match.
      const bool match =
          ((unsigned long long)key >> (shift + 8)) == (unsigned long long)prefix;
      if (match) atomicAdd(&hist[(key >> shift) & 0xFFu], 1u);  // ds_add_u32
    }
    __syncthreads();

    const unsigned c = hist[t];
    scan[t] = c;
    __syncthreads();
    for (int off = 1; off < THREADS; off <<= 1) {
      const unsigned v = (t >= off) ? scan[t - off] : 0u;
      __syncthreads();
      scan[t] += v;
      __syncthreads();
    }
    const unsigned incl = scan[t];
    const unsigned excl = incl - c;
    if (c > 0u && rank >= excl && rank < incl) {  // exactly one thread true
      sel_bin = (unsigned)t;
      sel_excl = excl;
    }
    __syncthreads();
    prefix = (prefix << 8) | sel_bin;
    rank -= sel_excl;
    __syncthreads();
  }

  if (t == 0) thr_bits[o] = prefix;  // exact bit pattern of sorted[RANK]
}

// ---------------------------------------------------------------------------
// Kernel 2: apply the mask over a 32(o) x 32(k) tile.
//   out[o,k] = |w[o,k]|, zeroed where |w|<=thr[o] && (eval || rand_u[k,o]<=0.5)
// rand_u (K,O-major) staged through LDS (async global->LDS when available),
// read back transposed with a conflict-free padded stride.
// ---------------------------------------------------------------------------
__global__ __launch_bounds__(256) void tw_mask_kernel(
    const float* __restrict__ w, const float* __restrict__ rand_u,
    const int* __restrict__ is_training, const unsigned* __restrict__ thr_bits,
    float* __restrict__ out) {
  __shared__ float ru[32][33];     // [k_in_tile][o_in_tile], padded (64 banks)
  __shared__ unsigned sthr[32];

  const int tx = threadIdx.x;      // lane within wave32
  const int ty = threadIdx.y;      // wave id (8 waves)
  const int k0 = blockIdx.x * 32;
  const int o0 = blockIdx.y * 32;

  if (ty == 0) sthr[tx] = thr_bits[o0 + tx];

#pragma unroll
  for (int rr = 0; rr < 4; ++rr) {
    const int r = ty + rr * 8;     // k row of the rand_u tile
    const float* g = rand_u + (size_t)(k0 + r) * O_DIM + (o0 + tx);  // coalesced in o
#if USE_ASYNC_LDS
    __builtin_amdgcn_global_load_async_to_lds_b32(
        (gas1_i32)const_cast<float*>(g), (gas3_i32)&ru[r][tx], 0, 0);
#else
    ru[r][tx] = *g;
#endif
  }
#if USE_ASYNC_LDS
  __builtin_amdgcn_s_wait_asynccnt(0);  // this wave's async ops complete
#endif
  __syncthreads();

  const int training = *is_training;

#pragma unroll
  for (int it = 0; it < 4; ++it) {
    const int i = ty + it * 8;                       // o offset in tile
    const size_t idx = (size_t)(o0 + i) * K_DIM + (k0 + tx);
    const float aval = fabsf(w[idx]);                // coalesced in k
    const unsigned key = __float_as_uint(aval);      // nonneg: uint order == float order
    const bool drop =
        (key <= sthr[i]) && (training == 0 || ru[tx][i] <= 0.5f);
    out[idx] = drop ? 0.0f : aval;                   // coalesced in k
  }
}

// ---------------------------------------------------------------------------
extern "C" void kernel_launch(void* const* d_in, const int* in_sizes, int n_in,
                              void* d_out, int out_size, void* d_ws, size_t ws_size,
                              hipStream_t stream) {
  (void)in_sizes; (void)n_in; (void)out_size; (void)ws_size;

  const float* w = (const float*)d_in[0];       // (O, C, KH, KW) = (O, K) flat
  const float* rand_u = (const float*)d_in[1];  // (K, O) flat
  const int* is_training = (const int*)d_in[2]; // scalar on device
  float* out = (float*)d_out;
  unsigned* thr_bits = (unsigned*)d_ws;         // O_DIM * 4 bytes of scratch

  tw_threshold_kernel<<<O_DIM, THREADS, 0, stream>>>(w, thr_bits);

  dim3 grid(K_DIM / 32, O_DIM / 32);            // 288 x 64 tiles
  dim3 block(32, 8);                            // 8 wave32s
  tw_mask_kernel<<<grid, block, 0, stream>>>(w, rand_u, is_training, thr_bits, out);
}